// MCUDetectionLoss_12610023981300
// MI455X (gfx1250) — compile-verified
//
#include <hip/hip_runtime.h>
#include <hip/hip_bf16.h>

typedef __attribute__((ext_vector_type(2))) float v2f;
typedef __attribute__((ext_vector_type(8))) float v8f;

#define NUM_CLASSES 80
#define NGT 32
#define TOPK 9
#define PI_F 3.14159265358979323846f

__device__ __forceinline__ float clampf(float x, float lo, float hi) {
    return fminf(fmaxf(x, lo), hi);
}
__device__ __forceinline__ float sigmoidf(float x) {
    return 1.0f / (1.0f + __expf(-x));
}
__device__ __forceinline__ float bce_logits(float x, float t) {
    // max(x,0) - x*t + log1p(exp(-|x|))
    return fmaxf(x, 0.0f) - x * t + log1pf(__expf(-fabsf(x)));
}

// ---------------------------------------------------------------------------
// Workspace init: assignment keys -> ~0 (so atomicMin works), partials -> 0
// ---------------------------------------------------------------------------
__global__ void init_ws_kernel(unsigned long long* __restrict__ assign, int n_assign,
                               float* __restrict__ partials, int n_part) {
    int i = blockIdx.x * blockDim.x + threadIdx.x;
    if (i < n_assign) assign[i] = 0xFFFFFFFFFFFFFFFFull;
    if (i < n_part)   partials[i] = 0.0f;
}

// ---------------------------------------------------------------------------
// SimOTALite assignment: one thread per (batch, gt).
// Top-9 nearest cells are provably inside the clipped 7x7 window around the
// containing cell. Contested cells resolved by u64 atomicMin on
// key = (bits(d2) << 32) | gt_index  (min-d2 wins, ties -> lowest gt index,
// matching jnp.argmin first-index tie-break).
// ---------------------------------------------------------------------------
__global__ void assign_kernel(const float* __restrict__ gt_boxes,
                              unsigned long long* __restrict__ assign,
                              int B, int H, int W) {
    int id = blockIdx.x * blockDim.x + threadIdx.x;
    if (id >= B * NGT) return;
    int b = id / NGT, n = id - b * NGT;

    float cx = gt_boxes[(b * NGT + n) * 4 + 0] * (float)W;
    float cy = gt_boxes[(b * NGT + n) * 4 + 1] * (float)H;
    int ix = min(max((int)floorf(cx), 0), W - 1);
    int iy = min(max((int)floorf(cy), 0), H - 1);

    float dist[49];
    int   flat[49];
    int cnt = 0;
    for (int dy = -3; dy <= 3; ++dy) {
        int yy = iy + dy;
        if (yy < 0 || yy >= H) continue;
        float fy = ((float)yy + 0.5f) - cy;
        float fy2 = fy * fy;
        for (int dx = -3; dx <= 3; ++dx) {
            int xx = ix + dx;
            if (xx < 0 || xx >= W) continue;
            float fx = ((float)xx + 0.5f) - cx;
            dist[cnt] = fy2 + fx * fx;
            flat[cnt] = yy * W + xx;
            ++cnt;
        }
    }
    // Clipped window is at least 4x4 = 16 cells >= TOPK.
    unsigned long long* base = assign + (size_t)b * H * W;
    for (int k = 0; k < TOPK; ++k) {
        int   best = -1;
        float bd = 3.4e38f;
        int   bf = 0x7fffffff;
        for (int i = 0; i < cnt; ++i) {
            if (flat[i] < 0) continue;
            if (dist[i] < bd || (dist[i] == bd && flat[i] < bf)) {
                bd = dist[i]; bf = flat[i]; best = i;
            }
        }
        unsigned long long key =
            ((unsigned long long)__float_as_uint(bd) << 32) | (unsigned int)n;
        atomicMin(base + bf, key);
        flat[best] = -1;  // mark used
    }
}

// ---------------------------------------------------------------------------
// Per-cell loss: obj BCE everywhere; CIoU bbox + 80-class focal gathered only
// at positive cells. Block-level LDS tree reduction -> per-block partials
// (deterministic; no float atomics).
// Partials layout: [0]=obj, [1]=bbox, [2]=cls, [3]=npos  (each length P)
// ---------------------------------------------------------------------------
__global__ void loss_kernel(const float* __restrict__ obj_pred,
                            const float* __restrict__ cls_pred,
                            const float* __restrict__ reg_pred,
                            const float* __restrict__ gt_boxes,
                            const int*   __restrict__ gt_cls,
                            const unsigned long long* __restrict__ assign,
                            float* __restrict__ partials, int partial_base, int P,
                            int B, int H, int W) {
    const int HW = H * W;
    const int t = blockIdx.x * blockDim.x + threadIdx.x;

    float obj_acc = 0.0f, bbox_acc = 0.0f, cls_acc = 0.0f, np_acc = 0.0f;

    if (t < B * HW) {
        int b = t / HW;
        int flat = t - b * HW;
        int y = flat / W;
        int x = flat - y * W;

        unsigned long long key = assign[(size_t)b * HW + flat];
        bool pos = (key != 0xFFFFFFFFFFFFFFFFull);
        float tgt = pos ? 1.0f : 0.0f;

        // objectness BCE over all cells
        float o = obj_pred[t];
        obj_acc = bce_logits(o, tgt);

        if (pos) {
            np_acc = 1.0f;
            int n = (int)(key & 0xFFFFFFFFu);

            // ---- bbox CIoU ----
            const size_t rb = (size_t)b * 4 * HW + flat;
            float r0 = reg_pred[rb + 0 * (size_t)HW];
            float r1 = reg_pred[rb + 1 * (size_t)HW];
            float r2 = reg_pred[rb + 2 * (size_t)HW];
            float r3 = reg_pred[rb + 3 * (size_t)HW];
            float invW = 1.0f / (float)W, invH = 1.0f / (float)H;
            float px = ((float)x + sigmoidf(r0)) * invW;
            float py = ((float)y + sigmoidf(r1)) * invH;
            float pw = __expf(clampf(r2, -4.0f, 4.0f)) * invW;
            float ph = __expf(clampf(r3, -4.0f, 4.0f)) * invH;

            float tx = gt_boxes[(b * NGT + n) * 4 + 0];
            float ty = gt_boxes[(b * NGT + n) * 4 + 1];
            float tw = gt_boxes[(b * NGT + n) * 4 + 2];
            float th = gt_boxes[(b * NGT + n) * 4 + 3];

            float px1 = px - pw * 0.5f, py1 = py - ph * 0.5f;
            float px2 = px + pw * 0.5f, py2 = py + ph * 0.5f;
            float tx1 = tx - tw * 0.5f, ty1 = ty - th * 0.5f;
            float tx2 = tx + tw * 0.5f, ty2 = ty + th * 0.5f;

            float iw = fmaxf(fminf(px2, tx2) - fmaxf(px1, tx1), 0.0f);
            float ih = fmaxf(fminf(py2, ty2) - fmaxf(py1, ty1), 0.0f);
            float inter = iw * ih;
            float uni = pw * ph + tw * th - inter + 1e-7f;
            float iou = inter / uni;
            float cd = (px - tx) * (px - tx) + (py - ty) * (py - ty);
            float ew = fmaxf(px2, tx2) - fminf(px1, tx1);
            float eh = fmaxf(py2, ty2) - fminf(py1, ty1);
            float c2 = ew * ew + eh * eh + 1e-7f;
            float dat = atanf(tw / (th + 1e-7f)) - atanf(pw / (ph + 1e-7f));
            float v = (4.0f / (PI_F * PI_F)) * dat * dat;
            float alpha = v / (1.0f - iou + v + 1e-7f);
            float ciou = clampf(iou - cd / c2 - alpha * v, -1.0f, 1.0f);
            bbox_acc = 1.0f - ciou;

            // ---- class focal loss (one-hot target) ----
            int gc = gt_cls[b * NGT + n];
            if (gc >= 0 && gc < NUM_CLASSES) {
                const float* cp = cls_pred + (size_t)b * NUM_CLASSES * HW + flat;
                // gfx1250 global_prefetch_b8: each channel is a distinct cacheline
                for (int c = 0; c < NUM_CLASSES; c += 8)
                    __builtin_prefetch(cp + (size_t)c * HW, 0, 0);
                float sum = 0.0f;
                for (int c = 0; c < NUM_CLASSES; ++c) {
                    float L = cp[(size_t)c * HW];
                    float tt = (c == gc) ? 1.0f : 0.0f;
                    L = clampf(L, -10.0f, 10.0f);
                    float p = clampf(sigmoidf(L), 1e-7f, 1.0f - 1e-7f);
                    float ce = clampf(bce_logits(L, tt), 0.0f, 100.0f);
                    float pt = p * tt + (1.0f - p) * (1.0f - tt);
                    float mod = (1.0f - pt) * (1.0f - pt);
                    float at = 0.25f * tt + 0.75f * (1.0f - tt);
                    sum += clampf(at * mod * ce, 0.0f, 100.0f);
                }
                cls_acc = sum;
            }
        }
    }

    // deterministic LDS tree reduction (256 threads)
    __shared__ float s0[256], s1[256], s2[256], s3[256];
    int tid = threadIdx.x;
    s0[tid] = obj_acc; s1[tid] = bbox_acc; s2[tid] = cls_acc; s3[tid] = np_acc;
    __syncthreads();
    for (int off = 128; off > 0; off >>= 1) {
        if (tid < off) {
            s0[tid] += s0[tid + off];
            s1[tid] += s1[tid + off];
            s2[tid] += s2[tid + off];
            s3[tid] += s3[tid + off];
        }
        __syncthreads();
    }
    if (tid == 0) {
        int pb = partial_base + blockIdx.x;
        partials[0 * P + pb] = s0[0];
        partials[1 * P + pb] = s1[0];
        partials[2 * P + pb] = s2[0];
        partials[3 * P + pb] = s3[0];
    }
}

// ---------------------------------------------------------------------------
// Finalize: single wave32. Reduce the 4 partial arrays (P each, P % 64 == 0)
// with V_WMMA_F32_16X16X4_F32: A = ones(16x4), B = 64-value chunk, C accum.
// D[m,n] = sum_k B[k,n] + C[m,n]  -> every row holds running column sums;
// summation order fixed in hardware -> deterministic.
// ---------------------------------------------------------------------------
__global__ void finalize_kernel(const float* __restrict__ partials, int P,
                                float* __restrict__ out, float obj_denom) {
    int lane = threadIdx.x;  // 0..31, EXEC all ones
    v2f ones = {1.0f, 1.0f};
    float sums[4];
#pragma unroll
    for (int a = 0; a < 4; ++a) {
        const float* pa = partials + (size_t)a * P;
        v8f c = {};
        for (int chunk = 0; chunk < P; chunk += 64) {
            v2f bm;
            bm.x = pa[chunk + lane];
            bm.y = pa[chunk + lane + 32];
            // (neg_a, A, neg_b, B, c_mod, C, reuse_a, reuse_b)
            c = __builtin_amdgcn_wmma_f32_16x16x4_f32(
                    false, ones, false, bm, (short)0, c, false, false);
        }
        // c[0] on lanes 0..15 = D[0, lane] (column sums). Sum the 16 columns:
        float v = c[0];
        v += __shfl_xor(v, 1, 32);
        v += __shfl_xor(v, 2, 32);
        v += __shfl_xor(v, 4, 32);
        v += __shfl_xor(v, 8, 32);
        sums[a] = v;
    }
    if (lane == 0) {
        float s_obj = sums[0], s_bbox = sums[1], s_cls = sums[2], s_np = sums[3];
        float inv = (s_np > 0.0f) ? (1.0f / fmaxf(s_np, 1.0f)) : 1.0f;
        float bbox = s_bbox * inv;
        float cls  = s_cls * inv;
        float obj  = s_obj / obj_denom;
        out[0] = bbox + obj + cls;  // total (weights all 1.0)
        out[1] = bbox;
        out[2] = obj;
        out[3] = cls;
    }
}

// ---------------------------------------------------------------------------
extern "C" void kernel_launch(void* const* d_in, const int* in_sizes, int n_in,
                              void* d_out, int out_size, void* d_ws, size_t ws_size,
                              hipStream_t stream) {
    const float* obj_p3   = (const float*)d_in[0];
    const float* cls_p3   = (const float*)d_in[1];
    const float* reg_p3   = (const float*)d_in[2];
    const float* obj_p4   = (const float*)d_in[3];
    const float* cls_p4   = (const float*)d_in[4];
    const float* reg_p4   = (const float*)d_in[5];
    const float* gt_boxes = (const float*)d_in[6];
    const int*   gt_cls   = (const int*)d_in[7];
    float* out = (float*)d_out;

    const int B = 16;
    const int H3 = 128, W3 = 128, H4 = 64, W4 = 64;
    const int n_as3 = B * H3 * W3;          // 262144 cells
    const int n_as4 = B * H4 * W4;          //  65536 cells
    const int n_assign = n_as3 + n_as4;     // 327680

    unsigned long long* as3 = (unsigned long long*)d_ws;
    unsigned long long* as4 = as3 + n_as3;
    float* partials = (float*)(as4 + n_as4);

    const int blocks3 = n_as3 / 256;        // 1024
    const int blocks4 = n_as4 / 256;        //  256
    const int P = blocks3 + blocks4;        // 1280 (multiple of 64)

    // 1) init workspace
    init_ws_kernel<<<(n_assign + 255) / 256, 256, 0, stream>>>(
        as3, n_assign, partials, 4 * P);

    // 2) GT -> cell assignment (both scales)
    assign_kernel<<<(B * NGT + 63) / 64, 64, 0, stream>>>(gt_boxes, as3, B, H3, W3);
    assign_kernel<<<(B * NGT + 63) / 64, 64, 0, stream>>>(gt_boxes, as4, B, H4, W4);

    // 3) per-cell losses -> block partials
    loss_kernel<<<blocks3, 256, 0, stream>>>(obj_p3, cls_p3, reg_p3, gt_boxes, gt_cls,
                                             as3, partials, 0, P, B, H3, W3);
    loss_kernel<<<blocks4, 256, 0, stream>>>(obj_p4, cls_p4, reg_p4, gt_boxes, gt_cls,
                                             as4, partials, blocks3, P, B, H4, W4);

    // 4) WMMA reduction + final scalars
    finalize_kernel<<<1, 32, 0, stream>>>(partials, P, out,
                                          (float)(B * (H3 * W3 + H4 * W4)));
}